// FeatureANI_80745385165456
// MI455X (gfx1250) — compile-verified
//
#include <hip/hip_runtime.h>
#include <cmath>

typedef float v2f __attribute__((ext_vector_type(2)));
typedef float v8f __attribute__((ext_vector_type(8)));

namespace {
constexpr int   kNcenters = 256 * 24;            // B*N = 6144
constexpr int   kN   = 24;
constexpr int   kKR  = 16;
constexpr int   kKA  = 32;
constexpr int   kNN  = kN - 1;                   // 23 neighbors
constexpr int   kNP  = kNN * (kNN - 1) / 2;      // 253 pairs
constexpr float kPi     = 3.14159265358979323846f;
constexpr float kRcr    = 5.2f;
constexpr float kRca    = 3.5f;
constexpr float kLog2e  = 1.4426950408889634f;
constexpr float kNegBig = -3.0e38f;              // finite "log2(0)" sentinel
}

#if __has_builtin(__builtin_amdgcn_wmma_f32_16x16x4_f32)
#define ANI_WMMA 1
#else
#define ANI_WMMA 0
#endif

// D = A(16x4, f32) x B(4x16, f32) + C(16x16, f32), wave32 distributed.
__device__ __forceinline__ v8f wmma4(v2f a, v2f b, v8f c) {
#if ANI_WMMA
  return __builtin_amdgcn_wmma_f32_16x16x4_f32(false, a, false, b, (short)0, c,
                                               false, false);
#else
  // Shuffle-based emulation of the documented VGPR layouts (compile fallback).
  const int lane = __lane_id();
  const int n  = lane & 15;
  const int mb = (lane < 16) ? 0 : 8;
  #pragma unroll
  for (int v = 0; v < 8; ++v) {
    float s = 0.f;
    #pragma unroll
    for (int k = 0; k < 4; ++k) {
      const int half = (k >= 2) ? 16 : 0;
      float av = __shfl((k & 1) ? a.y : a.x, (mb + v) + half, 32);
      float bv = __shfl((k & 1) ? b.y : b.x, n + half, 32);
      s = fmaf(av, bv, s);
    }
    c[v] += s;
  }
  return c;
#endif
}

__global__ __launch_bounds__(256)
void FeatureANI_kernel(const float* __restrict__ coords,
                       const int*   __restrict__ atom_types,
                       const float* __restrict__ EtaR,
                       const float* __restrict__ ShfR,
                       const float* __restrict__ Zeta,
                       const float* __restrict__ ShfZ,
                       const float* __restrict__ EtaA,
                       const float* __restrict__ ShfA,
                       float* __restrict__ out) {
  const int lane = threadIdx.x & 31;
  const int wave = threadIdx.x >> 5;
  const int c    = blockIdx.x * 8 + wave;        // center index, 0..6143
  const int b    = c / kN;
  const int i    = c % kN;
  const int tt   = atom_types[i];
  const bool lo  = (lane < 16);
  const int f0   = lane & 15;

  // ---- Phase 0: lane = neighbor. rij = x_i - x_j (skip-self indexing). ----
  const float cx = coords[(b * kN + i) * 3 + 0];
  const float cy = coords[(b * kN + i) * 3 + 1];
  const float cz = coords[(b * kN + i) * 3 + 2];

  int jj = (lane < kNN) ? lane : (kNN - 1);
  int j  = jj + (jj >= i);
  float rx = cx - coords[(b * kN + j) * 3 + 0];
  float ry = cy - coords[(b * kN + j) * 3 + 1];
  float rz = cz - coords[(b * kN + j) * 3 + 2];
  float d  = sqrtf(rx * rx + ry * ry + rz * rz);
  float fR = 0.5f * (__cosf(d * (kPi / kRcr)) + 1.0f);
  float fA = 0.5f * (__cosf(d * (kPi / kRca)) + 1.0f);

  // ---- Radial: lane = feature (both halves split 23 neighbors). ----
  const float etaRf = EtaR[tt * kKR + f0];
  const float shfRf = ShfR[tt * kKR + f0];
  float g2 = 0.f;
  #pragma unroll
  for (int tj = 0; tj < 12; ++tj) {
    int  jn = lo ? tj : (12 + tj);
    bool ok = (jn < kNN);
    jn = ok ? jn : (kNN - 1);
    float dj = __shfl(d,  jn, 32);
    float fj = __shfl(fR, jn, 32);
    fj = ok ? fj : 0.f;
    float u = dj - shfRf;
    g2 += __expf(-etaRf * u * u) * fj;
  }
  g2 += __shfl_xor(g2, 16, 32);

  // ---- Angular: per-lane feature params -> B matrices (4x16 f32). ----
  float zetaL[2];
  v2f Bcos[2], Bg[2];
  #pragma unroll
  for (int h = 0; h < 2; ++h) {
    const int f  = f0 + 16 * h;
    const float zt = Zeta[tt * kKA + f];
    const float sz = ShfZ[tt * kKA + f];
    const float ea = EtaA[tt * kKA + f];
    const float sa = ShfA[tt * kKA + f];
    zetaL[h] = zt;
    const float csa = cosf(sa), sna = sinf(sa);
    const float c0 = -ea * kLog2e;
    const float c1 = 2.0f * ea * sz * kLog2e;
    const float c2 = (1.0f - zt) - ea * sz * sz * kLog2e;
    // B(k,n): rows k=0,1 in lanes 0-15 (vgpr .x/.y), rows k=2,3 in lanes 16-31.
    Bcos[h].x = lo ? csa : 0.0f;
    Bcos[h].y = lo ? sna : 0.0f;
    Bg[h].x   = lo ? c0  : c2;
    Bg[h].y   = lo ? c1  : 1.0f;
  }

  // ---- Pair loop: 8 blocks of 32 pairs (lane = pair within block). ----
  v8f acc[2] = {};
  for (int t = 0; t < 8; ++t) {
    int  p     = t * 32 + lane;
    bool valid = (p < kNP);
    int  pc    = valid ? p : (kNP - 1);
    // Decode upper-triangular pair (a < b2) of the 23 neighbors.
    float disc = 2025.0f - 8.0f * (float)pc;     // (45-2a)^2 at row starts: exact
    int a = (int)floorf((45.0f - sqrtf(disc)) * 0.5f);
    a = (a < 0) ? 0 : ((a > kNN - 2) ? (kNN - 2) : a);
    int b2 = pc - (a * (45 - a)) / 2 + a + 1;

    float r12x = __shfl(rx, a, 32),  r12y = __shfl(ry, a, 32),  r12z = __shfl(rz, a, 32);
    float d12  = __shfl(d,  a, 32),  f12  = __shfl(fA, a, 32);
    float r13x = __shfl(rx, b2, 32), r13y = __shfl(ry, b2, 32), r13z = __shfl(rz, b2, 32);
    float d13  = __shfl(d,  b2, 32), f13  = __shfl(fA, b2, 32);

    float dotv = r12x * r13x + r12y * r13y + r12z * r13z;
    float cosv = dotv / (d12 * d13);
    cosv = fminf(fmaxf(cosv, -1.0f + 1e-7f), 1.0f - 1e-7f);
    float sinv = sqrtf(1.0f - cosv * cosv);
    float davg = 0.5f * (d12 + d13);
    float l2ff = fmaxf(__log2f(f12 * f13), kNegBig);
    l2ff = valid ? l2ff : kNegBig;               // kill padded pairs -> term 0

    // Cross-half swaps to build both 16-pair WMMA groups.
    float csw = __shfl_xor(cosv, 16, 32);
    float ssw = __shfl_xor(sinv, 16, 32);
    float dsw = __shfl_xor(davg, 16, 32);
    float lsw = __shfl_xor(l2ff, 16, 32);

    // A(m,k): k=0,1 at lanes 0-15 (row m=lane), k=2,3 at lanes 16-31 (m=lane-16).
    v2f Ac0, Ac1, Ag0, Ag1;
    Ac0.x = lo ? cosv : 0.0f;        Ac0.y = lo ? sinv : 0.0f;   // pairs 0..15
    Ac1.x = lo ? csw  : 0.0f;        Ac1.y = lo ? ssw  : 0.0f;   // pairs 16..31
    Ag0.x = lo ? davg * davg : 1.0f; Ag0.y = lo ? davg : lsw;
    Ag1.x = lo ? dsw * dsw   : 1.0f; Ag1.y = lo ? dsw  : l2ff;

    const v8f zero = {};
    #pragma unroll
    for (int g = 0; g < 2; ++g) {
      const v2f Ac = g ? Ac1 : Ac0;
      const v2f Ag = g ? Ag1 : Ag0;
      #pragma unroll
      for (int h = 0; h < 2; ++h) {
        v8f Dc = wmma4(Ac, Bcos[h], zero);   // cosang(pair, feat)
        v8f De = wmma4(Ag, Bg[h],  zero);    // full log2-exponent (pair, feat)
        const float zt = zetaL[h];
        #pragma unroll
        for (int v = 0; v < 8; ++v) {
          float u = fmaxf(1.0f + Dc[v], 0.0f);
          float e = fmaf(zt, __log2f(u), De[v]);
          acc[h][v] += __builtin_exp2f(e);
        }
      }
    }
  }

  // ---- Reduce D-layout accumulators (8 rows in-lane + half swap). ----
  float s0 = 0.f, s1 = 0.f;
  #pragma unroll
  for (int v = 0; v < 8; ++v) { s0 += acc[0][v]; s1 += acc[1][v]; }
  s0 += __shfl_xor(s0, 16, 32);
  s1 += __shfl_xor(s1, 16, 32);

  if (lo) {
    float* o = out + (size_t)c * (kKR + kKA);
    o[f0]      = g2;   // radial features 0..15
    o[16 + f0] = s0;   // angular features 0..15
    o[32 + f0] = s1;   // angular features 16..31
  }
}

extern "C" void kernel_launch(void* const* d_in, const int* in_sizes, int n_in,
                              void* d_out, int out_size, void* d_ws, size_t ws_size,
                              hipStream_t stream) {
  (void)in_sizes; (void)n_in; (void)out_size; (void)d_ws; (void)ws_size;
  const float* coords = (const float*)d_in[0];
  const int*   types  = (const int*)d_in[1];
  const float* EtaR   = (const float*)d_in[2];
  const float* ShfR   = (const float*)d_in[3];
  const float* Zeta   = (const float*)d_in[4];
  const float* ShfZ   = (const float*)d_in[5];
  const float* EtaA   = (const float*)d_in[6];
  const float* ShfA   = (const float*)d_in[7];
  float* out = (float*)d_out;

  dim3 grid(kNcenters / 8);   // 768 blocks, one wave32 per center
  dim3 block(256);            // 8 waves per block
  FeatureANI_kernel<<<grid, block, 0, stream>>>(coords, types, EtaR, ShfR,
                                                Zeta, ShfZ, EtaA, ShfA, out);
}